// MAF_44263932952824
// MI455X (gfx1250) — compile-verified
//
#include <hip/hip_runtime.h>
#include <math.h>

// Problem constants (match reference)
#define B_  4096
#define D_  64
#define H_  128
#define L_  3

typedef __attribute__((ext_vector_type(2))) float v2f;
typedef __attribute__((ext_vector_type(8))) float v8f;

// ---------------------------------------------------------------------------
// Workspace layout.
// Per net-slot s = layer*2 + r  (s = 0..5), FL_PER floats:
//   [    0,  8192)  W0t[k][u]   64 x 128  masked W0, columns sorted by m1
//   [ 8192, 24576)  W1t[u][g]  128 x 128  masked W1, rows m1-sorted, cols m2-sorted
//   [24576, 32768)  W2t[g][d]  128 x  64  masked W2, rows m2-sorted
//   [32768, 32896)  b0s[u]     bias0 in m1-sorted order
//   [32896, 33024)  b1s[g]     bias1 in m2-sorted order
// Int region per slot (INT_PER ints): pc1[64], pc2[64]
//   pc1[j] = #h1-units with degree m1 <= j  (degrees are in [1,63])
// ---------------------------------------------------------------------------
#define FL_PER  33024
#define INT_PER 128

// ===========================================================================
// Prep kernel: recover mask degrees, counting-sort hidden units by degree,
// scatter masked weights into sorted layout, build prefix-count tables.
// ===========================================================================
__global__ void maf_prep(const float* __restrict__ W0, const float* __restrict__ b0,
                         const float* __restrict__ W1, const float* __restrict__ b1,
                         const float* __restrict__ W2,
                         const float* __restrict__ M0, const float* __restrict__ M1,
                         const float* __restrict__ M2,
                         float* __restrict__ wsf, int* __restrict__ wsi) {
  const int s = blockIdx.x;     // 0..5
  const int h = threadIdx.x;    // 0..127
  const float* W0s = W0 + s*H_*D_;
  const float* M0s = M0 + s*H_*D_;
  const float* W1s = W1 + s*H_*H_;
  const float* M1s = M1 + s*H_*H_;
  const float* W2s = W2 + s*D_*H_;
  const float* M2s = M2 + s*D_*H_;
  float* W0t = wsf + s*FL_PER;
  float* W1t = W0t + 8192;
  float* W2t = W0t + 24576;
  float* b0s = W0t + 32768;
  float* b1s = W0t + 32896;
  int* pc1 = wsi + s*INT_PER;
  int* pc2 = pc1 + 64;

  __shared__ int deg1[H_], deg2[H_], rk2[H_];

  // degrees from masks: m1[h] = sum_k M0[h,k];  m2[g] = 64 - sum_d M2[d,g]
  float s1 = 0.f;
  for (int k = 0; k < D_; ++k) s1 += M0s[h*D_ + k];
  deg1[h] = (int)(s1 + 0.5f);
  float s2 = 0.f;
  for (int d = 0; d < D_; ++d) s2 += M2s[d*H_ + h];
  deg2[h] = D_ - (int)(s2 + 0.5f);
  __syncthreads();

  // stable ranks (counting sort), O(H) per thread
  const int d1 = deg1[h], d2 = deg2[h];
  int r1 = 0, r2 = 0;
  for (int p = 0; p < H_; ++p) {
    const int e1 = deg1[p]; r1 += (e1 < d1) || (e1 == d1 && p < h);
    const int e2 = deg2[p]; r2 += (e2 < d2) || (e2 == d2 && p < h);
  }
  rk2[h] = r2;
  __syncthreads();

  // scatter masked weights into degree-sorted layout
  for (int k = 0; k < D_; ++k)
    W0t[k*H_ + r1] = M0s[h*D_ + k] * W0s[h*D_ + k];
  b0s[r1] = b0[s*H_ + h];
  b1s[r2] = b1[s*H_ + h];
  for (int g = 0; g < H_; ++g)                 // h plays h1-orig; g is h2-orig
    W1t[r1*H_ + rk2[g]] = M1s[g*H_ + h] * W1s[g*H_ + h];
  for (int d = 0; d < D_; ++d)                 // h plays h2-orig
    W2t[r2*D_ + d] = M2s[d*H_ + h] * W2s[d*H_ + h];

  // prefix counts pc[j] = #units with degree <= j
  if (h < 64) {
    int c1 = 0, c2 = 0;
    for (int p = 0; p < H_; ++p) { c1 += (deg1[p] <= h); c2 += (deg2[p] <= h); }
    pc1[h] = c1; pc2[h] = c2;
  }
}

// ===========================================================================
// Main kernel: one wave = one 16-row batch tile, fully independent (no
// barriers).  Blocked-triangular schedule: dense off-diagonal contributions
// via V_WMMA_F32_16X16X4_F32, 16-step diagonal blocks via short VALU gathers.
// All weight-table pointers are direct offsets from the kernel-arg pointer so
// loads stay in the GLOBAL address space (LOADcnt only, no DScnt coupling).
// ===========================================================================
#define WAVES       4
#define SM_PER_WAVE 19456

__global__ __launch_bounds__(32*WAVES, 1)
void maf_main(const float* __restrict__ x_in, const float* __restrict__ b2,
              const int* __restrict__ perm,
              const float* __restrict__ wsf, const int* __restrict__ wsi,
              float* __restrict__ out) {
  extern __shared__ float sm[];
  const int wv   = threadIdx.x >> 5;
  const int lane = threadIdx.x & 31;
  const int row  = lane & 15;     // batch row within tile
  const int hf   = lane >> 4;     // wave half (0/1)

  float* S  = sm + wv * SM_PER_WAVE;
  float* XC = S;                  // [col][row]        64x16
  float* A1 = S + 1024;           // [net][u][row]   2x128x16
  float* T1 = S + 5120;
  float* A2 = S + 9216;
  float* T2 = S + 13312;
  float* AO = S + 17408;          // [net][d][row]    2x64x16

  const int tile  = blockIdx.x * WAVES + wv;
  const int rbase = tile * 16;

  // ---- load x tile, coalesced by row ----
  for (int rr = 0; rr < 16; ++rr) {
    XC[lane*16 + rr]        = x_in[(rbase + rr)*D_ + lane];
    XC[(lane + 32)*16 + rr] = x_in[(rbase + rr)*D_ + 32 + lane];
  }

  for (int layer = 0; layer < L_; ++layer) {
    // ---- column permutation: xc[:,c] = xc[:,perm[layer][c]] ----
    float tmp[32];
#pragma unroll
    for (int cc = 0; cc < 32; ++cc) {
      const int col = 2*cc + hf;
      tmp[cc] = XC[perm[layer*D_ + col]*16 + row];
    }
#pragma unroll
    for (int cc = 0; cc < 32; ++cc)
      XC[(2*cc + hf)*16 + row] = tmp[cc];

    // ---- per-net init + prefetch (unrolled: pointers constant-fold) ----
#pragma unroll
    for (int r = 0; r < 2; ++r) {
      const float* base = wsf + (layer*2 + r)*FL_PER;
      const float* B0s  = base + 32768;
      const float* B1s  = base + 32896;
      float* A1r = A1 + r*2048; float* A2r = A2 + r*2048; float* AOr = AO + r*1024;
      // warm the biggest weight table toward this WGP (global_prefetch_b8)
      for (int off = lane*64; off < H_*H_; off += 32*64)
        __builtin_prefetch(base + 8192 + off, 0, 1);
      for (int e = lane; e < 2048; e += 32) A1r[e] = B0s[e >> 4];
      for (int e = lane; e < 2048; e += 32) A2r[e] = B1s[e >> 4];
      for (int e = lane; e < 1024; e += 32) AOr[e] = b2[(layer*2 + r)*D_ + (e >> 4)];
    }

    for (int t = 0; t < 4; ++t) {
      // ================= block boundary: dense GEMMs (WMMA) =================
      if (t > 0) {
        const int kx0 = 16*(t-1);
#pragma unroll
        for (int r = 0; r < 2; ++r) {
          const float* Wb  = wsf + (layer*2 + r)*FL_PER;   // W0t
          const float* W1t = Wb + 8192;
          const float* W2t = Wb + 24576;
          const int*   pc1 = wsi + (layer*2 + r)*INT_PER;
          const int*   pc2 = pc1 + 64;
          float* A1r = A1 + r*2048; float* T1r = T1 + r*2048;
          float* A2r = A2 + r*2048; float* T2r = T2 + r*2048;
          float* AOr = AO + r*1024;
          const int prev1 = (t == 1) ? 0 : pc1[16*(t-1) - 1];
          const int new1  = pc1[16*t - 1];
          const int prev2 = (t == 1) ? 0 : pc2[16*(t-1) - 1];
          const int new2  = pc2[16*t - 1];

          // GEMM1: A1 += XC[:, kx0..kx0+15] @ W0t[kx0..kx0+15][:]   (K=16)
          for (int nt = 0; nt < 8; ++nt) {
            const int n0 = nt*16;
            v8f c;
#pragma unroll
            for (int q = 0; q < 8; ++q)
              c[q] = A1r[(n0 + row)*16 + q + 8*hf];
#pragma unroll
            for (int kc = 0; kc < 4; ++kc) {
              const int k0 = kx0 + 4*kc;
              v2f a, bm;
              a.x  = XC[(k0     + 2*hf)*16 + row];
              a.y  = XC[(k0 + 1 + 2*hf)*16 + row];
              bm.x = Wb[(k0     + 2*hf)*H_ + n0 + row];
              bm.y = Wb[(k0 + 1 + 2*hf)*H_ + n0 + row];
              c = __builtin_amdgcn_wmma_f32_16x16x4_f32(false, a, false, bm,
                                                        (short)0, c, false, false);
            }
#pragma unroll
            for (int q = 0; q < 8; ++q)
              A1r[(n0 + row)*16 + q + 8*hf] = c[q];
          }

          // GEMM2: A2 += T1[seg] @ W1t[seg][:]   (K = segment length, padded)
          for (int nt = 0; nt < 8; ++nt) {
            const int n0 = nt*16;
            v8f c;
#pragma unroll
            for (int q = 0; q < 8; ++q)
              c[q] = A2r[(n0 + row)*16 + q + 8*hf];
            for (int u0 = prev1; u0 < new1; u0 += 4) {
              const int ua = u0 + 2*hf, ub = u0 + 1 + 2*hf;
              v2f a, bm;
              a.x  = (ua < new1) ? T1r[ua*16 + row] : 0.f;
              a.y  = (ub < new1) ? T1r[ub*16 + row] : 0.f;
              bm.x = (ua < new1) ? W1t[ua*H_ + n0 + row] : 0.f;
              bm.y = (ub < new1) ? W1t[ub*H_ + n0 + row] : 0.f;
              c = __builtin_amdgcn_wmma_f32_16x16x4_f32(false, a, false, bm,
                                                        (short)0, c, false, false);
            }
#pragma unroll
            for (int q = 0; q < 8; ++q)
              A2r[(n0 + row)*16 + q + 8*hf] = c[q];
          }

          // GEMM3: AO += T2[seg] @ W2t[seg][:]
          for (int nt = 0; nt < 4; ++nt) {
            const int n0 = nt*16;
            v8f c;
#pragma unroll
            for (int q = 0; q < 8; ++q)
              c[q] = AOr[(n0 + row)*16 + q + 8*hf];
            for (int u0 = prev2; u0 < new2; u0 += 4) {
              const int ua = u0 + 2*hf, ub = u0 + 1 + 2*hf;
              v2f a, bm;
              a.x  = (ua < new2) ? T2r[ua*16 + row] : 0.f;
              a.y  = (ub < new2) ? T2r[ub*16 + row] : 0.f;
              bm.x = (ua < new2) ? W2t[ua*D_ + n0 + row] : 0.f;
              bm.y = (ub < new2) ? W2t[ub*D_ + n0 + row] : 0.f;
              c = __builtin_amdgcn_wmma_f32_16x16x4_f32(false, a, false, bm,
                                                        (short)0, c, false, false);
            }
#pragma unroll
            for (int q = 0; q < 8; ++q)
              AOr[(n0 + row)*16 + q + 8*hf] = c[q];
          }
        }
      }

      // ================= diagonal block: 16 sequential steps =================
      for (int jj = 0; jj < 16; ++jj) {
        const int j = 16*t + jj;
#pragma unroll
        for (int r = 0; r < 2; ++r) {
          const float* Wb  = wsf + (layer*2 + r)*FL_PER;   // W0t
          const float* W1t = Wb + 8192;
          const int*   pc1 = wsi + (layer*2 + r)*INT_PER;
          const int*   pc2 = pc1 + 64;
          float* A1r = A1 + r*2048; float* T1r = T1 + r*2048;
          float* A2r = A2 + r*2048; float* T2r = T2 + r*2048;
          // finalize h1 units with m1 == j  (prefix via GEMM1; in-block via VALU)
          const int e1lo  = (j == 0) ? 0 : pc1[j-1];
          const int e1hi  = pc1[j];
          const int base1 = (t == 0) ? 0 : pc1[16*t - 1];
          for (int u = e1lo + hf; u < e1hi; u += 2) {
            float v = A1r[u*16 + row];
            for (int k = 16*t; k < j; ++k)
              v += XC[k*16 + row] * Wb[k*H_ + u];
            T1r[u*16 + row] = tanhf(v);
          }
          // finalize h2 units with m2 == j
          const int e2lo = (j == 0) ? 0 : pc2[j-1];
          const int e2hi = pc2[j];
          const int p1j  = pc1[j];
          for (int g = e2lo + hf; g < e2hi; g += 2) {
            float v = A2r[g*16 + row];
            for (int u = base1; u < p1j; ++u)
              v += T1r[u*16 + row] * W1t[u*H_ + g];
            T2r[g*16 + row] = tanhf(v);
          }
        }
        // output column j:  mu (net0), alpha (net1)
        float acc[2];
#pragma unroll
        for (int r = 0; r < 2; ++r) {
          const float* W2t = wsf + (layer*2 + r)*FL_PER + 24576;
          const int*   pc2 = wsi + (layer*2 + r)*INT_PER + 64;
          const int base2 = (t == 0) ? 0 : pc2[16*t - 1];
          const int p2j   = pc2[j];
          const float* T2r = T2 + r*2048;
          float v = AO[r*1024 + j*16 + row];
          for (int g = base2; g < p2j; ++g)
            v += T2r[g*16 + row] * W2t[g*D_ + j];
          acc[r] = v;
        }
        const float sig = 0.1f + 1.9f / (1.f + __expf(-acc[1]));
        const float xj  = acc[0] + sig * XC[j*16 + row];
        if (hf == 0) XC[j*16 + row] = xj;
      }
    }
  }

  // ---- store result ----
  for (int rr = 0; rr < 16; ++rr) {
    out[(rbase + rr)*D_ + lane]      = XC[lane*16 + rr];
    out[(rbase + rr)*D_ + 32 + lane] = XC[(lane + 32)*16 + rr];
  }
}

// ===========================================================================
extern "C" void kernel_launch(void* const* d_in, const int* in_sizes, int n_in,
                              void* d_out, int out_size, void* d_ws, size_t ws_size,
                              hipStream_t stream) {
  const float* x   = (const float*)d_in[0];
  const float* W0  = (const float*)d_in[1];
  const float* b0  = (const float*)d_in[2];
  const float* W1  = (const float*)d_in[3];
  const float* b1  = (const float*)d_in[4];
  const float* W2  = (const float*)d_in[5];
  const float* b2  = (const float*)d_in[6];
  const float* M0  = (const float*)d_in[7];
  const float* M1  = (const float*)d_in[8];
  const float* M2  = (const float*)d_in[9];
  const int*   pm  = (const int*)  d_in[10];
  float* out = (float*)d_out;

  float* wsf = (float*)d_ws;
  int*   wsi = (int*)((char*)d_ws + (size_t)(L_*2) * FL_PER * sizeof(float));

  // 1) build degree-sorted masked weight tables + prefix counts
  maf_prep<<<L_*2, H_, 0, stream>>>(W0, b0, W1, b1, W2, M0, M1, M2, wsf, wsi);

  // 2) blocked-triangular MAF forward: 1 wave per 16-row tile, 4 waves/WG
  const int grid = B_ / (16 * WAVES);                       // 64 workgroups
  const size_t smem = (size_t)WAVES * SM_PER_WAVE * sizeof(float);  // ~304 KB
  maf_main<<<grid, 32*WAVES, smem, stream>>>(x, b2, pm, wsf, wsi, out);
}